// AVES_49323404427432
// MI455X (gfx1250) — compile-verified
//
#include <hip/hip_runtime.h>
#include <hip/hip_bf16.h>

#define NB 2048
#define NV 4000
#define NJ 25
#define NK 24
#define NBETA 15
#define NTILE (NV / 16)          // 250 16-vertex tiles
#define NSTEP 7                  // K padded 25 -> 28

typedef __attribute__((ext_vector_type(2))) float v2f;
typedef __attribute__((ext_vector_type(8))) float v8f;

// rot = [[0,1,0],[0,0,-1],[-1,0,0]];  x' = y, y' = -z, z' = -x
__device__ __forceinline__ void rot_read(const float* __restrict__ J, int j,
                                         float& x, float& y, float& z) {
    x =  J[j * 3 + 1];
    y = -J[j * 3 + 2];
    z = -J[j * 3 + 0];
}

// ---------------------------------------------------------------------------
// Kernel A: pack skinning weights into per-lane WMMA A-fragment layout.
// apack[t][s][lane] = { W[t*16 + l][k0], W[t*16 + l][k0+1] },  k0 = 4s + 2h,
// zero-padded for joints >= 25.  (l = lane&15, h = lane>>4)
// ---------------------------------------------------------------------------
__global__ __launch_bounds__(224) void apack_kernel(
    const float* __restrict__ W, float* __restrict__ apack)
{
    const int t = blockIdx.x;            // tile 0..249
    const int r = threadIdx.x;           // 0..223 = 7 steps * 32 lanes
    const int s = r >> 5, lane = r & 31;
    const int l = lane & 15, h = lane >> 4;
    const int k0 = 4 * s + 2 * h;
    const int v = t * 16 + l;
    v2f o;
    o.x = (k0     < NJ) ? W[(size_t)v * NJ + k0]     : 0.f;
    o.y = (k0 + 1 < NJ) ? W[(size_t)v * NJ + k0 + 1] : 0.f;
    ((v2f*)apack)[((size_t)t * NSTEP + s) * 32 + lane] = o;
}

// ---------------------------------------------------------------------------
// Kernel 1: forward kinematics. One thread per batch element.
// Produces trel[b][j][12] row-major [r00 r01 r02 t0' | r10.. | r20..] with
// t' = t_rel - R_rel * (scale * Jt[j]).  Tree: parents[i] = (i-1)/2.
// ---------------------------------------------------------------------------
__global__ __launch_bounds__(256) void fk_kernel(
    const float* __restrict__ Jraw, const float* __restrict__ go,
    const float* __restrict__ pose, const float* __restrict__ bone,
    const int* __restrict__ scale_i, float* __restrict__ trel)
{
    const int b = blockIdx.x * blockDim.x + threadIdx.x;
    if (b >= NB) return;
    const float scl = (float)scale_i[0];

    float TrR[12][9];   // only joints 0..11 can be parents
    float Trt[12][3];

    {   // j = 0 (root): R = global_orient, kin = scl * 1 * Jt[0]
        float r[9];
#pragma unroll
        for (int c = 0; c < 9; ++c) r[c] = go[(size_t)b * 9 + c];
        float jx, jy, jz;
        rot_read(Jraw, 0, jx, jy, jz);
        const float tx = scl * jx, ty = scl * jy, tz = scl * jz;
#pragma unroll
        for (int c = 0; c < 9; ++c) TrR[0][c] = r[c];
        Trt[0][0] = tx; Trt[0][1] = ty; Trt[0][2] = tz;
        const float cx = scl * jx, cy = scl * jy, cz = scl * jz;
        float* o = trel + ((size_t)b * NJ + 0) * 12;
        o[0] = r[0]; o[1] = r[1]; o[2]  = r[2]; o[3]  = tx - (r[0]*cx + r[1]*cy + r[2]*cz);
        o[4] = r[3]; o[5] = r[4]; o[6]  = r[5]; o[7]  = ty - (r[3]*cx + r[4]*cy + r[5]*cz);
        o[8] = r[6]; o[9] = r[7]; o[10] = r[8]; o[11] = tz - (r[6]*cx + r[7]*cy + r[8]*cz);
    }

#pragma unroll
    for (int j = 1; j < NJ; ++j) {
        const int p = (j - 1) >> 1;              // compile-time after unroll
        float r[9];
#pragma unroll
        for (int c = 0; c < 9; ++c)
            r[c] = pose[((size_t)b * (NJ - 1) + (j - 1)) * 9 + c];

        float jx, jy, jz, px, py, pz;
        rot_read(Jraw, j, jx, jy, jz);
        rot_read(Jraw, p, px, py, pz);
        const float bf = scl * bone[(size_t)b * (NJ - 1) + (j - 1)];
        const float kx = bf * (jx - px), ky = bf * (jy - py), kz = bf * (jz - pz);

        float Rc[9], tc[3];
#pragma unroll
        for (int i = 0; i < 3; ++i) {
#pragma unroll
            for (int k = 0; k < 3; ++k)
                Rc[3 * i + k] = TrR[p][3 * i + 0] * r[0 + k]
                              + TrR[p][3 * i + 1] * r[3 + k]
                              + TrR[p][3 * i + 2] * r[6 + k];
            tc[i] = TrR[p][3 * i + 0] * kx + TrR[p][3 * i + 1] * ky
                  + TrR[p][3 * i + 2] * kz + Trt[p][i];
        }
        if (j < 12) {
#pragma unroll
            for (int c = 0; c < 9; ++c) TrR[j][c] = Rc[c];
            Trt[j][0] = tc[0]; Trt[j][1] = tc[1]; Trt[j][2] = tc[2];
        }
        const float cx = scl * jx, cy = scl * jy, cz = scl * jz;
        float* o = trel + ((size_t)b * NJ + j) * 12;
        o[0] = Rc[0]; o[1] = Rc[1]; o[2]  = Rc[2]; o[3]  = tc[0] - (Rc[0]*cx + Rc[1]*cy + Rc[2]*cz);
        o[4] = Rc[3]; o[5] = Rc[4]; o[6]  = Rc[5]; o[7]  = tc[1] - (Rc[3]*cx + Rc[4]*cy + Rc[5]*cz);
        o[8] = Rc[6]; o[9] = Rc[7]; o[10] = Rc[8]; o[11] = tc[2] - (Rc[6]*cx + Rc[7]*cy + Rc[8]*cz);
    }
}

// ---------------------------------------------------------------------------
// Kernel B: pack per-batch T_rel into per-lane WMMA B-fragment layout.
// bpack[b][s][lane] = { Trel[b][k0][l], Trel[b][k0+1][l] }, zero for
// joints >= 25 or cols >= 12.
// ---------------------------------------------------------------------------
__global__ __launch_bounds__(224) void bpack_kernel(
    const float* __restrict__ trel, float* __restrict__ bpack)
{
    const int b = blockIdx.x;
    const int r = threadIdx.x;           // 0..223
    const int s = r >> 5, lane = r & 31;
    const int l = lane & 15, h = lane >> 4;
    const int k0 = 4 * s + 2 * h;
    const float* tb = trel + (size_t)b * NJ * 12;
    v2f o;
    o.x = (k0     < NJ && l < 12) ? tb[k0 * 12 + l]       : 0.f;
    o.y = (k0 + 1 < NJ && l < 12) ? tb[(k0 + 1) * 12 + l] : 0.f;
    ((v2f*)bpack)[((size_t)b * NSTEP + s) * 32 + lane] = o;
}

// ---------------------------------------------------------------------------
// Kernel 2: WMMA skinning. Block = 256 threads = 8 waves; each wave handles
// 32 vertices (two 16x16 C tiles) of one batch element.  Hot loop is
// branch-free: 3 coalesced global_load_b64 + 2 v_wmma_f32_16x16x4_f32.
// Then fused shape-blend + affine apply through LDS.
// ---------------------------------------------------------------------------
__global__ __launch_bounds__(256) void skin_kernel(
    const float* __restrict__ Vraw, const float* __restrict__ Bs,
    const float* __restrict__ apack, const float* __restrict__ betas,
    const float* __restrict__ transl, const float* __restrict__ bpack,
    const int* __restrict__ scale_i, float* __restrict__ out)
{
    const int b    = blockIdx.y;
    const int wave = threadIdx.x >> 5;
    const int lane = threadIdx.x & 31;
    const int l    = lane & 15;
    const int h    = lane >> 4;
    const int job  = blockIdx.x * 8 + wave;      // 0..124 valid
    const bool active = (job < NV / 32);

    __shared__ float Tb[8][32][13];  // per wave: 32 verts x 12 comps (+pad)
    __shared__ float sb[16];
    if (threadIdx.x < NBETA) sb[threadIdx.x] = betas[(size_t)b * NBETA + threadIdx.x];

    v8f c0 = {0.f, 0.f, 0.f, 0.f, 0.f, 0.f, 0.f, 0.f};
    v8f c1 = {0.f, 0.f, 0.f, 0.f, 0.f, 0.f, 0.f, 0.f};
    const int v0 = job * 32;

    if (active) {
        const v2f* A0 = (const v2f*)apack + ((size_t)(2 * job)     * NSTEP) * 32 + lane;
        const v2f* A1 = (const v2f*)apack + ((size_t)(2 * job + 1) * NSTEP) * 32 + lane;
        const v2f* Bp = (const v2f*)bpack + ((size_t)b * NSTEP) * 32 + lane;
#pragma unroll
        for (int s = 0; s < NSTEP; ++s) {
            const v2f a0 = A0[s * 32];
            const v2f a1 = A1[s * 32];
            const v2f bf = Bp[s * 32];
            // D = A(16x4) * B(4x16) + C ; fp32 WMMA, full precision
            c0 = __builtin_amdgcn_wmma_f32_16x16x4_f32(false, a0, false, bf,
                                                       (short)0, c0, false, false);
            c1 = __builtin_amdgcn_wmma_f32_16x16x4_f32(false, a1, false, bf,
                                                       (short)0, c1, false, false);
        }
        // C layout: VGPR r, lanes 0-15 -> (M=r, N=l); lanes 16-31 -> (M=8+r, N=l)
        if (l < 12) {
#pragma unroll
            for (int r = 0; r < 8; ++r) {
                Tb[wave][8 * h + r][l]      = c0[r];
                Tb[wave][16 + 8 * h + r][l] = c1[r];
            }
        }
    }
    __syncthreads();

    if (active) {
        const float scl = (float)scale_i[0];
        const int v = v0 + lane;
        // V = scale * rot(V_t)  +  sum_k betas[k] * B_shape[k] (B_shape NOT rotated)
        float px =  scl * Vraw[(size_t)v * 3 + 1];
        float py = -scl * Vraw[(size_t)v * 3 + 2];
        float pz = -scl * Vraw[(size_t)v * 3 + 0];
#pragma unroll
        for (int k = 0; k < NBETA; ++k) {
            const float co = sb[k];
            const float* bp = Bs + ((size_t)k * NV + v) * 3;
            px = fmaf(co, bp[0], px);
            py = fmaf(co, bp[1], py);
            pz = fmaf(co, bp[2], pz);
        }
        const float* m = Tb[wave][lane];   // blended 3x4 affine
        const float tx = transl[(size_t)b * 3 + 0];
        const float ty = transl[(size_t)b * 3 + 1];
        const float tz = transl[(size_t)b * 3 + 2];
        const float ox = m[0] * px + m[1] * py + m[2]  * pz + m[3]  + tx;
        const float oy = m[4] * px + m[5] * py + m[6]  * pz + m[7]  + ty;
        const float oz = m[8] * px + m[9] * py + m[10] * pz + m[11] + tz;
        float* op = out + ((size_t)b * NV + v) * 3;
        op[0] = ox; op[1] = oy; op[2] = oz;
    }
}

// ---------------------------------------------------------------------------
// Kernel 3: keypoints = vert2kpt[24 x 4000] @ verts[b][4000 x 3].
// One block per batch element; shuffle + LDS reduction, direct final store.
// ---------------------------------------------------------------------------
__global__ __launch_bounds__(256) void kpt_kernel(
    const float* __restrict__ v2k, const float* __restrict__ verts,
    float* __restrict__ kout)
{
    const int b = blockIdx.x;
    float acc[NK][3];
#pragma unroll
    for (int k = 0; k < NK; ++k) { acc[k][0] = 0.f; acc[k][1] = 0.f; acc[k][2] = 0.f; }

    for (int n = threadIdx.x; n < NV; n += 256) {
        const float* vp = verts + ((size_t)b * NV + n) * 3;
        const float vx = vp[0], vy = vp[1], vz = vp[2];
#pragma unroll
        for (int k = 0; k < NK; ++k) {
            const float w = v2k[(size_t)k * NV + n];
            acc[k][0] = fmaf(w, vx, acc[k][0]);
            acc[k][1] = fmaf(w, vy, acc[k][1]);
            acc[k][2] = fmaf(w, vz, acc[k][2]);
        }
    }

    __shared__ float red[8][NK * 3];
    const int wv = threadIdx.x >> 5;
    const int ln = threadIdx.x & 31;
#pragma unroll
    for (int k = 0; k < NK; ++k) {
#pragma unroll
        for (int c = 0; c < 3; ++c) {
            float v = acc[k][c];
            v += __shfl_down(v, 16, 32);
            v += __shfl_down(v,  8, 32);
            v += __shfl_down(v,  4, 32);
            v += __shfl_down(v,  2, 32);
            v += __shfl_down(v,  1, 32);
            if (ln == 0) red[wv][k * 3 + c] = v;
        }
    }
    __syncthreads();
    if (threadIdx.x < NK * 3) {
        float s = 0.f;
#pragma unroll
        for (int w = 0; w < 8; ++w) s += red[w][threadIdx.x];
        kout[(size_t)b * (NK * 3) + threadIdx.x] = s;
    }
}

// ---------------------------------------------------------------------------
extern "C" void kernel_launch(void* const* d_in, const int* in_sizes, int n_in,
                              void* d_out, int out_size, void* d_ws, size_t ws_size,
                              hipStream_t stream)
{
    (void)in_sizes; (void)n_in; (void)out_size; (void)ws_size;
    const float* V_t    = (const float*)d_in[0];   // [4000,3]
    const float* J_t    = (const float*)d_in[1];   // [25,3]
    const float* Bs     = (const float*)d_in[2];   // [15,4000,3]
    const float* W      = (const float*)d_in[3];   // [4000,25]
    const float* v2k    = (const float*)d_in[4];   // [24,4000]
    const float* go     = (const float*)d_in[5];   // [2048,1,3,3]
    const float* pose   = (const float*)d_in[6];   // [2048,24,3,3]
    const float* bone   = (const float*)d_in[7];   // [2048,24]
    const float* betas  = (const float*)d_in[8];   // [2048,15]
    const float* transl = (const float*)d_in[9];   // [2048,3]
    const int*   scale  = (const int*)d_in[11];    // scalar
    float* out = (float*)d_out;                    // verts then keypoints

    // workspace layout (floats): trel | bpack | apack
    float* trel  = (float*)d_ws;                             // 2048*25*12  = 614400
    float* bpack = trel + (size_t)NB * NJ * 12;              // 2048*7*32*2 = 917504
    float* apack = bpack + (size_t)NB * NSTEP * 32 * 2;      // 250*7*32*2  = 112000

    apack_kernel<<<dim3(NTILE), dim3(224), 0, stream>>>(W, apack);
    fk_kernel<<<dim3(NB / 256), dim3(256), 0, stream>>>(J_t, go, pose, bone, scale, trel);
    bpack_kernel<<<dim3(NB), dim3(224), 0, stream>>>(trel, bpack);
    skin_kernel<<<dim3(16, NB), dim3(256), 0, stream>>>(V_t, Bs, apack, betas, transl,
                                                        bpack, scale, out);
    kpt_kernel<<<dim3(NB), dim3(256), 0, stream>>>(v2k, out, out + (size_t)NB * NV * 3);
}